// ElementwiseSparsity_9105330668182
// MI455X (gfx1250) — compile-verified
//
#include <hip/hip_runtime.h>

// Problem dims (from reference): x[B,M,T], We[H,M], be[H], Wc[M,H], bc[M]
#define Bsz  4
#define Mdim 512
#define Tdim 8192
#define Hdim 2048
#define KEEP 64

#define BM 64      // H-rows per block tile
#define BN 128     // T-cols per block tile
#define BK 32      // K slice
#define LDA 36     // sA row stride (floats): mult of 4 (16B align), conflict-free frag reads
#define LDB 136    // sB row stride (floats): mult of 4, 136 mod 64 = 8 -> conflict-free frag reads
#define NBINS 4096
#define CAND  4096

typedef float v2f __attribute__((ext_vector_type(2)));
typedef float v8f __attribute__((ext_vector_type(8)));
typedef int   v4i __attribute__((ext_vector_type(4)));

typedef __attribute__((address_space(1))) v4i* g_v4i_ptr;
typedef __attribute__((address_space(3))) v4i* l_v4i_ptr;

// ---- workspace layout (u32 units) ----
#define WS_HIST 0                      // Bsz*NBINS
#define WS_CNT  (WS_HIST + Bsz*NBINS)  // Bsz
#define WS_TH   (WS_CNT + Bsz)         // Bsz
#define WS_ZERO (WS_TH + Bsz)          // zero [0, WS_ZERO)
#define WS_TVAL (WS_ZERO + 8)          // Bsz*KEEP floats
#define WS_TIDX (WS_TVAL + Bsz*KEEP)   // Bsz*KEEP u32
#define WS_CVAL (WS_TIDX + Bsz*KEEP)   // Bsz*CAND floats
#define WS_CIDX (WS_CVAL + Bsz*CAND)   // Bsz*CAND u32

#if defined(__has_builtin)
#if __has_builtin(__builtin_amdgcn_global_load_async_to_lds_b128)
#define HAVE_ASYNC_LDS 1
#endif
#if __has_builtin(__builtin_amdgcn_s_wait_asynccnt)
#define HAVE_WAIT_ASYNC 1
#endif
#endif
#ifndef HAVE_ASYNC_LDS
#define HAVE_ASYNC_LDS 0
#endif
#ifndef HAVE_WAIT_ASYNC
#define HAVE_WAIT_ASYNC 0
#endif

__device__ __forceinline__ void copy16_g2lds(const float* gsrc, float* ldst) {
#if HAVE_ASYNC_LDS
  // param0: addrspace(1) v4i*  (global source)
  // param1: addrspace(3) v4i*  (LDS destination)
  __builtin_amdgcn_global_load_async_to_lds_b128(
      (g_v4i_ptr)(unsigned long long)gsrc,
      (l_v4i_ptr)(unsigned int)(unsigned long long)ldst,
      0, 0);
#else
  *(float4*)ldst = *(const float4*)gsrc;
#endif
}

__device__ __forceinline__ void wait_lds_copies() {
#if HAVE_ASYNC_LDS
#if HAVE_WAIT_ASYNC
  __builtin_amdgcn_s_wait_asynccnt(0);
#else
  asm volatile("s_wait_asynccnt 0" ::: "memory");
#endif
#endif
}

// monotonic float->u32 key (larger float => larger key)
__device__ __forceinline__ unsigned int fmono(float f) {
  unsigned int u = __float_as_uint(f);
  return u ^ (((unsigned int)((int)u >> 31)) | 0x80000000u);
}

// ---------------------------------------------------------------------------
// Pass 0: zero histogram / counters
__global__ void zero_ws_kernel(unsigned int* ws) {
  for (int i = blockIdx.x * blockDim.x + threadIdx.x; i < WS_ZERO;
       i += gridDim.x * blockDim.x)
    ws[i] = 0u;
}

// ---------------------------------------------------------------------------
// Pass 1: h = We @ x + be via v_wmma_f32_16x16x4_f32; write h to scratch
// (sparse region of d_out) and build per-batch histogram of fmono keys.
__global__ void __launch_bounds__(256)
gemm_hist_kernel(const float* __restrict__ x, const float* __restrict__ We,
                 const float* __restrict__ be, float* __restrict__ hscr,
                 unsigned int* __restrict__ ghist) {
  __shared__ float sA[BM * LDA];
  __shared__ float sB[BK * LDB];
  __shared__ unsigned int sHist[NBINS];

  const int tid = threadIdx.x;
  const int lane = tid & 31;
  const int wave = tid >> 5;
  const int wi = wave >> 2;   // 0..1 : 32-row group
  const int wj = wave & 3;    // 0..3 : 32-col group
  const int lh = lane >> 4;   // half-wave select
  const int ll = lane & 15;

  const int b = blockIdx.z;
  const int hBase = blockIdx.y * BM;
  const int tBase = blockIdx.x * BN;
  const size_t HT = (size_t)Hdim * Tdim;
  const size_t xbase = (size_t)b * Mdim * Tdim;

  for (int i = tid; i < NBINS; i += 256) sHist[i] = 0u;

  v8f acc[2][2] = {};

  for (int kb = 0; kb < Mdim; kb += BK) {
    __syncthreads();  // previous tile fully consumed
    // We tile: 64 rows x 32 floats  (512 16B-chunks, 2 per thread)
#pragma unroll
    for (int i = 0; i < 2; ++i) {
      int c = tid + i * 256;
      int r = c >> 3, c4 = (c & 7) * 4;
      copy16_g2lds(We + (size_t)(hBase + r) * Mdim + kb + c4, &sA[r * LDA + c4]);
    }
    // x tile: 32 rows x 128 floats (1024 16B-chunks, 4 per thread)
#pragma unroll
    for (int i = 0; i < 4; ++i) {
      int c = tid + i * 256;
      int r = c >> 5, c4 = (c & 31) * 4;
      copy16_g2lds(x + xbase + (size_t)(kb + r) * Tdim + tBase + c4,
                   &sB[r * LDB + c4]);
    }
    wait_lds_copies();
    __syncthreads();

#pragma unroll
    for (int kk = 0; kk < BK; kk += 4) {
      const int k0 = kk + (lh << 1);  // lanes 0-15: K=kk,kk+1 ; 16-31: kk+2,kk+3
      const int ra0 = (wi * 32 + ll) * LDA + k0;
      const int ra1 = (wi * 32 + 16 + ll) * LDA + k0;
      const int c0 = wj * 32 + ll;
      v2f a0 = {sA[ra0], sA[ra0 + 1]};
      v2f a1 = {sA[ra1], sA[ra1 + 1]};
      v2f b0 = {sB[k0 * LDB + c0], sB[(k0 + 1) * LDB + c0]};
      v2f b1 = {sB[k0 * LDB + c0 + 16], sB[(k0 + 1) * LDB + c0 + 16]};
      acc[0][0] = __builtin_amdgcn_wmma_f32_16x16x4_f32(false, a0, false, b0,
                                                        (short)0, acc[0][0], false, false);
      acc[0][1] = __builtin_amdgcn_wmma_f32_16x16x4_f32(false, a0, false, b1,
                                                        (short)0, acc[0][1], false, false);
      acc[1][0] = __builtin_amdgcn_wmma_f32_16x16x4_f32(false, a1, false, b0,
                                                        (short)0, acc[1][0], false, false);
      acc[1][1] = __builtin_amdgcn_wmma_f32_16x16x4_f32(false, a1, false, b1,
                                                        (short)0, acc[1][1], false, false);
    }
  }

  // Epilogue: bias, scratch store, histogram
  float* hb = hscr + (size_t)b * HT;
#pragma unroll
  for (int ti = 0; ti < 2; ++ti) {
#pragma unroll
    for (int tj = 0; tj < 2; ++tj) {
      const int gRow0 = hBase + wi * 32 + ti * 16 + (lh << 3);
      const int gCol = tBase + wj * 32 + tj * 16 + ll;
#pragma unroll
      for (int v = 0; v < 8; ++v) {
        const int gRow = gRow0 + v;
        float val = acc[ti][tj][v] + be[gRow];
        hb[(size_t)gRow * Tdim + gCol] = val;
        atomicAdd(&sHist[fmono(val) >> 20], 1u);
      }
    }
  }

  __syncthreads();
  for (int i = tid; i < NBINS; i += 256) {
    unsigned int c = sHist[i];
    if (c) atomicAdd(&ghist[b * NBINS + i], c);
  }
}

// ---------------------------------------------------------------------------
// Pass 2: per-batch threshold = lower edge of bin where top-down cumsum >= KEEP
__global__ void thresh_kernel(const unsigned int* __restrict__ hist,
                              unsigned int* __restrict__ thresh) {
  int b = threadIdx.x;
  if (b >= Bsz) return;
  unsigned long long c = 0;
  int bin = NBINS - 1;
  for (; bin > 0; --bin) {
    c += hist[b * NBINS + bin];
    if (c >= KEEP) break;
  }
  thresh[b] = ((unsigned int)bin) << 20;
}

// ---------------------------------------------------------------------------
// Pass 3: stream h scratch, collect candidates >= threshold
__global__ void collect_kernel(const float* __restrict__ hscr,
                               const unsigned int* __restrict__ thresh,
                               unsigned int* __restrict__ cnt,
                               float* __restrict__ cval,
                               unsigned int* __restrict__ cidx) {
  const size_t HTQ = (size_t)Hdim * Tdim / 4;
  const size_t NQ = (size_t)Bsz * HTQ;
  for (size_t q = (size_t)blockIdx.x * blockDim.x + threadIdx.x; q < NQ;
       q += (size_t)gridDim.x * blockDim.x) {
    const int b = (int)(q / HTQ);
    const unsigned int th = thresh[b];
    const unsigned int r0 = (unsigned int)((q % HTQ) * 4);
    float4 v = ((const float4*)hscr)[q];
    float e[4] = {v.x, v.y, v.z, v.w};
#pragma unroll
    for (int j = 0; j < 4; ++j) {
      if (fmono(e[j]) >= th) {
        unsigned int pos = atomicAdd(&cnt[b], 1u);
        if (pos < CAND) {
          cval[b * CAND + pos] = e[j];
          cidx[b * CAND + pos] = r0 + j;
        }
      }
    }
  }
}

// ---------------------------------------------------------------------------
// Pass 4: exact top-KEEP per batch from candidates (one block per batch)
__global__ void __launch_bounds__(256)
topk_kernel(const unsigned int* __restrict__ cnt, const float* __restrict__ cval,
            const unsigned int* __restrict__ cidx, float* __restrict__ tval,
            unsigned int* __restrict__ tidx) {
  __shared__ float sv[CAND];
  __shared__ unsigned int si[CAND];
  __shared__ float rv[256];
  __shared__ int ri[256];
  const int b = blockIdx.x, tid = threadIdx.x;
  int n = (int)cnt[b];
  if (n > CAND) n = CAND;
  for (int i = tid; i < CAND; i += 256) {
    if (i < n) {
      sv[i] = cval[b * CAND + i];
      si[i] = cidx[b * CAND + i];
    } else {
      sv[i] = -__builtin_inff();
      si[i] = 0u;
    }
  }
  __syncthreads();
  for (int k = 0; k < KEEP; ++k) {
    float bv = -__builtin_inff();
    int bi = 0;
    for (int i = tid; i < CAND; i += 256)
      if (sv[i] > bv) { bv = sv[i]; bi = i; }
    rv[tid] = bv;
    ri[tid] = bi;
    __syncthreads();
    for (int s = 128; s > 0; s >>= 1) {
      if (tid < s && rv[tid + s] > rv[tid]) {
        rv[tid] = rv[tid + s];
        ri[tid] = ri[tid + s];
      }
      __syncthreads();
    }
    if (tid == 0) {
      int j = ri[0];
      tval[b * KEEP + k] = rv[0];
      tidx[b * KEEP + k] = si[j];
      sv[j] = -__builtin_inff();
    }
    __syncthreads();
  }
}

// ---------------------------------------------------------------------------
// Pass 5: out = bc broadcast; sparse = 0 (overwrites the h scratch)
__global__ void init_out_kernel(float* __restrict__ dst,
                                const float* __restrict__ bc) {
  const size_t OUTQ = (size_t)Bsz * Mdim * Tdim / 4;
  const size_t TOTQ = OUTQ + (size_t)Bsz * Hdim * Tdim / 4;
  for (size_t q = (size_t)blockIdx.x * blockDim.x + threadIdx.x; q < TOTQ;
       q += (size_t)gridDim.x * blockDim.x) {
    float4 w;
    if (q < OUTQ) {
      int m = (int)(((q * 4) / Tdim) % Mdim);
      float v = bc[m];
      w = make_float4(v, v, v, v);
    } else {
      w = make_float4(0.f, 0.f, 0.f, 0.f);
    }
    ((float4*)dst)[q] = w;
  }
}

// ---------------------------------------------------------------------------
// Pass 6: scatter KEEP values into sparse; rank-1 column updates into out.
__global__ void scatter_kernel(float* __restrict__ out, float* __restrict__ sp,
                               const float* __restrict__ Wc,
                               const float* __restrict__ tval,
                               const unsigned int* __restrict__ tidx) {
  const int blk = blockIdx.x;
  const int b = blk >> 6;      // / KEEP
  const int k = blk & 63;      // % KEEP
  const float val = tval[b * KEEP + k];
  const unsigned int idx = tidx[b * KEEP + k];
  const int h = (int)(idx / Tdim);
  const int t = (int)(idx % Tdim);
  if (threadIdx.x == 0) sp[(size_t)b * Hdim * Tdim + idx] = val;
  for (int m = threadIdx.x; m < Mdim; m += blockDim.x)
    atomicAdd(&out[((size_t)b * Mdim + m) * Tdim + t],
              Wc[(size_t)m * Hdim + h] * val);
}

// ---------------------------------------------------------------------------
extern "C" void kernel_launch(void* const* d_in, const int* in_sizes, int n_in,
                              void* d_out, int out_size, void* d_ws, size_t ws_size,
                              hipStream_t stream) {
  const float* x  = (const float*)d_in[0];
  const float* We = (const float*)d_in[1];
  const float* be = (const float*)d_in[2];
  const float* Wc = (const float*)d_in[3];
  const float* bc = (const float*)d_in[4];

  float* out = (float*)d_out;
  float* sparse = out + (size_t)Bsz * Mdim * Tdim;  // also h scratch
  unsigned int* ws = (unsigned int*)d_ws;

  zero_ws_kernel<<<64, 256, 0, stream>>>(ws);

  dim3 g1(Tdim / BN, Hdim / BM, Bsz);
  gemm_hist_kernel<<<g1, 256, 0, stream>>>(x, We, be, sparse, ws + WS_HIST);

  thresh_kernel<<<1, 64, 0, stream>>>(ws + WS_HIST, ws + WS_TH);

  collect_kernel<<<4096, 256, 0, stream>>>(sparse, ws + WS_TH, ws + WS_CNT,
                                           (float*)(ws + WS_CVAL), ws + WS_CIDX);

  topk_kernel<<<Bsz, 256, 0, stream>>>(ws + WS_CNT, (const float*)(ws + WS_CVAL),
                                       ws + WS_CIDX, (float*)(ws + WS_TVAL),
                                       ws + WS_TIDX);

  init_out_kernel<<<8192, 256, 0, stream>>>(out, bc);

  scatter_kernel<<<Bsz * KEEP, 256, 0, stream>>>(out, sparse, Wc,
                                                 (const float*)(ws + WS_TVAL),
                                                 ws + WS_TIDX);
}